// GCN_79242146611680
// MI455X (gfx1250) — compile-verified
//
#include <hip/hip_runtime.h>
#include <hip/hip_bf16.h>

typedef float v2f __attribute__((ext_vector_type(2)));
typedef float v8f __attribute__((ext_vector_type(8)));

#define BN_EPS 1e-5f

// ---------------------------------------------------------------------------
// Degree / normalization precompute (shared across all 5 GCN layers)
// ---------------------------------------------------------------------------
__global__ void k_deg_init(float* __restrict__ deg, int n) {
  int i = blockIdx.x * blockDim.x + threadIdx.x;
  if (i < n) deg[i] = 1.0f;  // self-loop contributes weight 1 to every node
}

__global__ void k_deg_accum(const int* __restrict__ dst, const float* __restrict__ w,
                            float* __restrict__ deg, int e) {
  int i = blockIdx.x * blockDim.x + threadIdx.x;
  if (i < e) atomicAdd(&deg[dst[i]], w[i]);
}

__global__ void k_dinv(float* __restrict__ deg, int n) {
  int i = blockIdx.x * blockDim.x + threadIdx.x;
  if (i < n) {
    float d = deg[i];
    deg[i] = (d > 0.f) ? rsqrtf(fmaxf(d, 1e-12f)) : 0.f;
  }
}

__global__ void k_norm(const int* __restrict__ src, const int* __restrict__ dst,
                       const float* __restrict__ w, const float* __restrict__ dinv,
                       float* __restrict__ norm, int e) {
  int i = blockIdx.x * blockDim.x + threadIdx.x;
  if (i < e) norm[i] = dinv[src[i]] * w[i] * dinv[dst[i]];
}

// ---------------------------------------------------------------------------
// Dense GEMM D[M,Nc] = A[M,K] @ B[K,Nc] via V_WMMA_F32_16X16X4_F32 (fp32 exact)
// One wave per 16x16 output tile; K stepped by 4.
// A 16x4 layout: lane l holds M=l%16, K = 2*(l>=16)+j in vgpr j.
// B 4x16 layout: lane l holds N=l%16, K = 2*(l>=16)+j in vgpr j.
// C/D: vgpr r, lane l -> (M = r + 8*(l>=16), N = l%16).
// ---------------------------------------------------------------------------
__global__ void k_gemm_wmma_f32(const float* __restrict__ A, const float* __restrict__ B,
                                float* __restrict__ D, int M, int K, int Nc) {
  int wave = (blockIdx.x * blockDim.x + threadIdx.x) >> 5;
  int lane = threadIdx.x & 31;
  int ntile = (Nc + 15) >> 4;
  int mtile = M >> 4;                       // M is a multiple of 16 (N=100000)
  if (wave >= mtile * ntile) return;        // wave-uniform exit: EXEC stays all-1s
  int tm = wave / ntile;
  int tn = wave - tm * ntile;

  int half = lane >> 4;                     // 0: low half-wave, 1: high half-wave
  int l16  = lane & 15;
  int col  = tn * 16 + l16;
  int colc = (col < Nc) ? col : 0;          // clamp (select, no branch)
  float bmask = (col < Nc) ? 1.f : 0.f;     // zero OOB columns (last MLP layer, Nc=10)

  const float* arow = A + (size_t)(tm * 16 + l16) * K + 2 * half;
  const float* bp   = B + (size_t)(2 * half) * Nc + colc;

  v8f acc = {};
  for (int k0 = 0; k0 < K; k0 += 4) {
    v2f a = *(const v2f*)(arow + k0);                 // A[m][k0+2*half .. +1]
    v2f b;
    b.x = bp[(size_t)k0 * Nc] * bmask;                // B[k0+2*half][col]
    b.y = bp[(size_t)(k0 + 1) * Nc] * bmask;          // B[k0+2*half+1][col]
    acc = __builtin_amdgcn_wmma_f32_16x16x4_f32(
        /*neg_a=*/false, a, /*neg_b=*/false, b,
        /*c_mod=*/(short)0, acc, /*reuse_a=*/false, /*reuse_b=*/false);
  }

  if (col < Nc) {
    float* drow = D + (size_t)(tm * 16 + 8 * half) * Nc + col;
#pragma unroll
    for (int r = 0; r < 8; ++r) drow[(size_t)r * Nc] = acc[r];
  }
}

// ---------------------------------------------------------------------------
// agg = dinv^2 * xt   (self-loop term; doubles as zero-free initialization)
// i indexes float4 elements; qlog = log2(d/4)
// ---------------------------------------------------------------------------
__global__ void k_selfloop_init(const float* __restrict__ xt, const float* __restrict__ dinv,
                                float* __restrict__ agg, int n, int qlog) {
  int i = blockIdx.x * blockDim.x + threadIdx.x;
  int total = n << qlog;
  if (i >= total) return;
  int row = i >> qlog;
  float s = dinv[row];
  s = s * s;
  float4 v = ((const float4*)xt)[i];
  ((float4*)agg)[i] = make_float4(v.x * s, v.y * s, v.z * s, v.w * s);
}

// ---------------------------------------------------------------------------
// Edge scatter: agg[dst] += norm[e] * xt[src], float4-vectorized gather,
// scalar f32 atomics (global_atomic_add_f32). qlog = log2(d/4).
// ---------------------------------------------------------------------------
__global__ void k_edge_scatter(const float* __restrict__ xt, const int* __restrict__ src,
                               const int* __restrict__ dst, const float* __restrict__ norm,
                               float* __restrict__ agg, int e, int qlog) {
  int i = blockIdx.x * blockDim.x + threadIdx.x;
  int total = e << qlog;
  if (i >= total) return;
  int ei = i >> qlog;
  int c4 = (i & ((1 << qlog) - 1)) << 2;
  int dlog = qlog + 2;
  int s = src[ei];
  int t = dst[ei];
  float nv = norm[ei];
  float4 v = *(const float4*)(xt + ((size_t)s << dlog) + c4);
  float* ap = agg + ((size_t)t << dlog) + c4;
  atomicAdd(ap + 0, nv * v.x);
  atomicAdd(ap + 1, nv * v.y);
  atomicAdd(ap + 2, nv * v.z);
  atomicAdd(ap + 3, nv * v.w);
}

// ---------------------------------------------------------------------------
// BatchNorm (training-mode batch stats, biased var) + ReLU
// ---------------------------------------------------------------------------
__global__ void k_zero(float* __restrict__ p, int n) {
  int i = blockIdx.x * blockDim.x + threadIdx.x;
  if (i < n) p[i] = 0.f;
}

__global__ void k_bn_stats(const float* __restrict__ h, float* __restrict__ stats,
                           int n, int d) {
  int c = threadIdx.x;
  if (c >= d) return;
  int rows = (n + gridDim.x - 1) / gridDim.x;
  int r0 = blockIdx.x * rows;
  int r1 = min(n, r0 + rows);
  float s = 0.f, s2 = 0.f;
  for (int r = r0; r < r1; ++r) {
    float v = h[(size_t)r * d + c];
    s += v;
    s2 += v * v;
  }
  atomicAdd(&stats[c], s);
  atomicAdd(&stats[d + c], s2);
}

__global__ void k_bn_relu(const float* __restrict__ in, float* __restrict__ out,
                          const float* __restrict__ stats,
                          const float* __restrict__ gamma, const float* __restrict__ beta,
                          int n, int d, float inv_n) {
  int i = blockIdx.x * blockDim.x + threadIdx.x;
  if (i >= n * d) return;
  int c = i % d;
  float mean = stats[c] * inv_n;
  float var = stats[d + c] * inv_n - mean * mean;
  float y = (in[i] - mean) * rsqrtf(var + BN_EPS) * gamma[c] + beta[c];
  out[i] = fmaxf(y, 0.f);
}

// ---------------------------------------------------------------------------
// Host-side orchestration
// ---------------------------------------------------------------------------
static inline int cdiv(long long a, int b) { return (int)((a + b - 1) / b); }

extern "C" void kernel_launch(void* const* d_in, const int* in_sizes, int n_in,
                              void* d_out, int out_size, void* d_ws, size_t ws_size,
                              hipStream_t stream) {
  (void)in_sizes; (void)n_in; (void)out_size; (void)ws_size;

  const int N = 100000;
  const int E = 800000;

  const float* x       = (const float*)d_in[0];
  const int*   src     = (const int*)d_in[1];        // edge_index row 0
  const int*   dst     = ((const int*)d_in[1]) + E;  // edge_index row 1
  const float* ew      = (const float*)d_in[2];

  const float* gcn_w[5];     for (int j = 0; j < 5; ++j) gcn_w[j]     = (const float*)d_in[3 + j];
  // gcn_b (d_in[8..12]) cancels exactly in training-mode BN -> dropped (all zeros anyway)
  const float* gcn_gamma[5]; for (int j = 0; j < 5; ++j) gcn_gamma[j] = (const float*)d_in[13 + j];
  const float* gcn_beta[5];  for (int j = 0; j < 5; ++j) gcn_beta[j]  = (const float*)d_in[18 + j];
  const float* mlp_w[3];     for (int j = 0; j < 3; ++j) mlp_w[j]     = (const float*)d_in[23 + j];
  // mlp_b (d_in[26..28]) also cancels in BN
  const float* mlp_gamma[3]; for (int j = 0; j < 3; ++j) mlp_gamma[j] = (const float*)d_in[29 + j];
  const float* mlp_beta[3];  for (int j = 0; j < 3; ++j) mlp_beta[j]  = (const float*)d_in[32 + j];

  // Workspace carve-up
  float* ws = (float*)d_ws;
  float* buf0 = ws;                          // xt / pre-BN activations  [N*256]
  float* buf1 = buf0 + (size_t)N * 256;      // agg / post-BN h          [N*256]
  float* dinv = buf1 + (size_t)N * 256;      // [N]  (deg -> dinv in place)
  float* norm = dinv + N;                    // [E]
  float* stats = norm + E;                   // [512] column sum / sumsq

  const int TB = 256;

  // ---- degree / norm (once, reused by all GCN layers) ----
  k_deg_init<<<cdiv(N, TB), TB, 0, stream>>>(dinv, N);
  k_deg_accum<<<cdiv(E, TB), TB, 0, stream>>>(dst, ew, dinv, E);
  k_dinv<<<cdiv(N, TB), TB, 0, stream>>>(dinv, N);
  k_norm<<<cdiv(E, TB), TB, 0, stream>>>(src, dst, ew, dinv, norm, E);

  // ---- 5 GCN layers: gemm -> selfloop-init -> edge scatter -> BN+ReLU ----
  const int gdi[5] = {64, 256, 128, 64, 32};
  const int gdo[5] = {256, 128, 64, 32, 16};
  const float* h = x;  // layer-0 input
  for (int l = 0; l < 5; ++l) {
    int K = gdi[l], d = gdo[l];
    int qlog = __builtin_ctz(d) - 2;  // log2(d/4), d is a power of two here

    int waves = (N >> 4) * ((d + 15) >> 4);
    k_gemm_wmma_f32<<<cdiv((long long)waves * 32, TB), TB, 0, stream>>>(h, gcn_w[l], buf0, N, K, d);

    k_selfloop_init<<<cdiv((long long)N << qlog, TB), TB, 0, stream>>>(buf0, dinv, buf1, N, qlog);
    k_edge_scatter<<<cdiv((long long)E << qlog, TB), TB, 0, stream>>>(buf0, src, dst, norm, buf1, E, qlog);

    k_zero<<<2, TB, 0, stream>>>(stats, 2 * d);
    k_bn_stats<<<256, ((d + 31) & ~31), 0, stream>>>(buf1, stats, N, d);
    k_bn_relu<<<cdiv((long long)N * d, TB), TB, 0, stream>>>(buf1, buf1, stats, gcn_gamma[l],
                                                             gcn_beta[l], N, d, 1.f / (float)N);
    h = buf1;
  }

  // ---- 3 MLP layers: gemm -> BN+ReLU ----
  const int mdi[3] = {16, 32, 32};
  const int mdo[3] = {32, 32, 10};
  for (int l = 0; l < 3; ++l) {
    int K = mdi[l], d = mdo[l];

    int waves = (N >> 4) * ((d + 15) >> 4);
    k_gemm_wmma_f32<<<cdiv((long long)waves * 32, TB), TB, 0, stream>>>(h, mlp_w[l], buf0, N, K, d);

    k_zero<<<2, TB, 0, stream>>>(stats, 2 * d);
    k_bn_stats<<<256, ((d + 31) & ~31), 0, stream>>>(buf0, stats, N, d);

    float* outp = (l == 2) ? (float*)d_out : buf1;
    k_bn_relu<<<cdiv((long long)N * d, TB), TB, 0, stream>>>(buf0, outp, stats, mlp_gamma[l],
                                                             mlp_beta[l], N, d, 1.f / (float)N);
    h = buf1;
  }
}